// ConnectedLoss_83760452206646
// MI455X (gfx1250) — compile-verified
//
#include <hip/hip_runtime.h>
#include <stdint.h>

// ---------------- problem constants ----------------
#define Bn   4
#define Cn   3
#define Hn   512
#define Wn   512
#define HW   (Hn * Wn)            // 262144 == 2^18
#define NPIX (Bn * HW)            // 1048576
#define NT   5
#define MAXC 512
#define LOG2F_ 0.69314718055994530942f

typedef float v2f __attribute__((ext_vector_type(2)));
typedef float v8f __attribute__((ext_vector_type(8)));

// ---------------- workspace arena (dword offsets) ----------------
#define OFF_BASES 0      // 4 floats: sum_bce, sum_p, sum_pt, sum_t
#define OFF_TCNT  4      // 8 ints: target histogram (5 used)
#define OFF_PFLAG 12     // 1 int: bitmask of pred_masks values present
#define OFF_NCNT  16     // 2 ints: component counters for v=1,2
#define OFF_V0    32
// per-v stats block layout (relative):
#define SA    0                       // A[MAXC]
#define SS1   (MAXC)                  // S1[MAXC]
#define SNF   (2*MAXC)                // N[MAXC]
#define SBX   (3*MAXC)                // Bx[MAXC*5]
#define SSG   (3*MAXC + 5*MAXC)      // Sg[MAXC*5]
#define SCF   (3*MAXC + 10*MAXC)     // Cnt[MAXC*5]
#define SLOSS (3*MAXC + 15*MAXC)     // loss[MAXC*5]
#define VSTR  (3*MAXC + 20*MAXC)     // 11776 dwords per v
#define ARENA_DW (32 + 2*VSTR)

// byte offsets inside d_ws
#define WS_PM     0u
#define WS_LABELS (1u << 20)
#define WS_CID    (5u << 20)
#define WS_ARENA  (9u << 20)

// ---------------- helpers ----------------
__device__ __forceinline__ float sigmoidf_(float x) { return 1.0f / (1.0f + expf(-x)); }

// WMMA f32 reduction step: D += A(16x4 data) * B(4x16 ones) -> per-row sums, exact f32.
__device__ __forceinline__ v8f wmma_red(v2f a, v8f c) {
#if __has_builtin(__builtin_amdgcn_wmma_f32_16x16x4_f32)
  const v2f ones = {1.0f, 1.0f};
  return __builtin_amdgcn_wmma_f32_16x16x4_f32(false, a, false, ones, (short)0, c,
                                               false, false);
#else
  c[0] += a.x + a.y;   // VALU fallback (histogram will show wmma==0 -> iterate)
  return c;
#endif
}

__device__ __forceinline__ float wave_total(v8f c) {
  float s = c[0] + c[1] + c[2] + c[3] + c[4] + c[5] + c[6] + c[7];
#if __has_builtin(__builtin_amdgcn_wmma_f32_16x16x4_f32)
  // total = s(lane) + s(lane^16): D rows m in VGPRs0-7 (lanes0-15) / m+8 (lanes16-31)
  s += __shfl_xor(s, 16, 32);
#else
  for (int o = 16; o; o >>= 1) s += __shfl_xor(s, o, 32);
#endif
  return s;
}

// ---------------- kernel: zero arena ----------------
__global__ void k_zero(float* a) {
  int i = blockIdx.x * blockDim.x + threadIdx.x;
  if (i < ARENA_DW) a[i] = 0.0f;
}

// ---------------- kernel: argmax + base BCE-Dice sums via WMMA ----------------
__global__ __launch_bounds__(256) void k_argmax_base(const float* __restrict__ pred,
                                                     const int* __restrict__ tgt,
                                                     uint8_t* __restrict__ pm,
                                                     float* arenaF, int* arenaI) {
  const int lane = threadIdx.x & 31;
  const int wave = (blockIdx.x * blockDim.x + threadIdx.x) >> 5;
  const int nw   = (gridDim.x * blockDim.x) >> 5;

  v8f ab = {}, ap = {}, apt = {}, at = {};
  int pfl = 0;
  float tc[NT] = {0.f, 0.f, 0.f, 0.f, 0.f};

  for (int base = wave * 64; base < NPIX; base += nw * 64) {
    float qb[2], qp[2], qpt[2], qt[2];
#pragma unroll
    for (int j = 0; j < 2; ++j) {
      int p   = base + j * 32 + lane;
      int b   = p >> 18;
      int rem = p & (HW - 1);
      const float* pc = pred + (size_t)b * Cn * HW + rem;
      float c0 = pc[0], c1 = pc[HW], c2 = pc[2 * HW];
      int m = 0; float best = c0;
      if (c1 > best) { m = 1; best = c1; }
      if (c2 > best) { m = 2; }
      pm[p] = (uint8_t)m;
      pfl |= 1 << m;
      int t = tgt[p]; t = t < 0 ? 0 : (t >= NT ? NT - 1 : t);
      tc[t] += 1.0f;
      float tb = (t > 0) ? 1.0f : 0.0f;
      float x  = (m > 0) ? c1 : 0.0f;            // masked logit
      float sg = sigmoidf_(x);
      qb[j]  = fmaxf(x, 0.0f) - x * tb + log1pf(expf(-fabsf(x)));
      qp[j]  = sg;
      qpt[j] = sg * tb;
      qt[j]  = tb;
    }
    v2f vb  = {qb[0],  qb[1]},  vp = {qp[0],  qp[1]};
    v2f vpt = {qpt[0], qpt[1]}, vt = {qt[0],  qt[1]};
    ab  = wmma_red(vb,  ab);
    ap  = wmma_red(vp,  ap);
    apt = wmma_red(vpt, apt);
    at  = wmma_red(vt,  at);
  }

  float sb = wave_total(ab), sp = wave_total(ap);
  float st = wave_total(at), spt = wave_total(apt);
  for (int o = 16; o; o >>= 1) pfl |= __shfl_xor(pfl, o, 32);
#pragma unroll
  for (int t = 0; t < NT; ++t)
    for (int o = 16; o; o >>= 1) tc[t] += __shfl_xor(tc[t], o, 32);

  if (lane == 0) {
    atomicAdd(&arenaF[OFF_BASES + 0], sb);
    atomicAdd(&arenaF[OFF_BASES + 1], sp);
    atomicAdd(&arenaF[OFF_BASES + 2], spt);
    atomicAdd(&arenaF[OFF_BASES + 3], st);
    atomicOr(&arenaI[OFF_PFLAG], pfl);
    for (int t = 0; t < NT; ++t) atomicAdd(&arenaI[OFF_TCNT + t], (int)tc[t]);
  }
}

// ---------------- connected components: atomic min union-find ----------------
__device__ __forceinline__ int cc_find(const int* L, int a) {
  int p = L[a];
  while (p != a) { a = p; p = L[a]; }
  return a;
}
__device__ void cc_union(int* L, int a, int b) {
  a = cc_find(L, a); b = cc_find(L, b);
  while (a != b) {
    if (a > b) { int t = a; a = b; b = t; }   // a < b: hook b under a (min-root)
    int old = atomicMin(&L[b], a);
    if (old == b) break;
    a = cc_find(L, a); b = cc_find(L, old);
  }
}

__global__ void k_cc_init(const uint8_t* pm, int* L, int v) {
  int p = blockIdx.x * blockDim.x + threadIdx.x;
  if (p < NPIX) L[p] = (pm[p] == (uint8_t)v) ? p : -1;
}

__global__ void k_cc_merge(const uint8_t* pm, int* L, int v) {
  int p = blockIdx.x * blockDim.x + threadIdx.x;
  if (p >= NPIX) return;
  if (pm[p] != (uint8_t)v) return;
  int rem = p & (HW - 1);
  int h = rem >> 9, w = rem & (Wn - 1);
  if (w + 1 < Wn && pm[p + 1]  == (uint8_t)v) cc_union(L, p, p + 1);
  if (h + 1 < Hn && pm[p + Wn] == (uint8_t)v) cc_union(L, p, p + Wn);
}

__global__ void k_cc_flatten(int* L) {
  int p = blockIdx.x * blockDim.x + threadIdx.x;
  if (p < NPIX && L[p] >= 0) L[p] = cc_find(L, p);
}

__global__ void k_cc_compact(const uint8_t* pm, const int* L, int* cid, int* counter, int v) {
  int p = blockIdx.x * blockDim.x + threadIdx.x;
  if (p >= NPIX) return;
  if (pm[p] == (uint8_t)v && L[p] == p) cid[p] = 1 + atomicAdd(counter, 1);
}

// ---------------- per-component / per-(f,t) statistics (LDS-binned) ----------------
__global__ __launch_bounds__(256) void k_stats(const float* __restrict__ pred,
                                               const int* __restrict__ tgt,
                                               const uint8_t* __restrict__ pm,
                                               const int* __restrict__ L,
                                               const int* __restrict__ cid,
                                               int v, float* st) {
  __shared__ float sA[3 * MAXC];        // A | S1 | N
  __shared__ float sP[3 * MAXC * NT];   // Bx | Sg | Cnt
  for (int i = threadIdx.x; i < 3 * MAXC; i += blockDim.x) sA[i] = 0.0f;
  for (int i = threadIdx.x; i < 3 * MAXC * NT; i += blockDim.x) sP[i] = 0.0f;
  __syncthreads();

  for (int p = blockIdx.x * blockDim.x + threadIdx.x; p < NPIX;
       p += gridDim.x * blockDim.x) {
    int f = 0;
    if (pm[p] == (uint8_t)v) {
      int r = L[p];
      f = cid[r];
      if (f < 0) f = 0;
      if (f > MAXC - 1) f = MAXC - 1;
    }
    int b = p >> 18, rem = p & (HW - 1);
    float x  = pred[(size_t)(b * Cn + v) * HW + rem];
    int t = tgt[p]; t = t < 0 ? 0 : (t >= NT ? NT - 1 : t);
    float sg = sigmoidf_(x);
    float a  = fmaxf(x, 0.0f) + log1pf(expf(-fabsf(x))) - LOG2F_;
    atomicAdd(&sA[f], a);
    atomicAdd(&sA[MAXC + f], sg);
    atomicAdd(&sA[2 * MAXC + f], 1.0f);
    int ft = f * NT + t;
    atomicAdd(&sP[ft], x);
    atomicAdd(&sP[MAXC * NT + ft], sg);
    atomicAdd(&sP[2 * MAXC * NT + ft], 1.0f);
  }
  __syncthreads();

  for (int i = threadIdx.x; i < MAXC; i += blockDim.x) {
    if (sA[i] != 0.f)            atomicAdd(&st[SA + i],  sA[i]);
    if (sA[MAXC + i] != 0.f)     atomicAdd(&st[SS1 + i], sA[MAXC + i]);
    if (sA[2 * MAXC + i] != 0.f) atomicAdd(&st[SNF + i], sA[2 * MAXC + i]);
  }
  for (int i = threadIdx.x; i < MAXC * NT; i += blockDim.x) {
    if (sP[i] != 0.f)                 atomicAdd(&st[SBX + i], sP[i]);
    if (sP[MAXC * NT + i] != 0.f)     atomicAdd(&st[SSG + i], sP[MAXC * NT + i]);
    if (sP[2 * MAXC * NT + i] != 0.f) atomicAdd(&st[SCF + i], sP[2 * MAXC * NT + i]);
  }
}

// ---------------- pair loss table: loss[f][t] from decomposed stats ----------------
__global__ void k_pair(float* st, const int* Tcnt) {
  int i = blockIdx.x * blockDim.x + threadIdx.x;
  if (i >= MAXC * NT) return;
  int f = i / NT, t = i % NT;
  float Tt = (float)Tcnt[t];
  const float Nf = (float)NPIX;
  float bce   = (st[SA + f] - st[SBX + i]) / Nf + LOG2F_;
  float inter = st[SSG + i] + 0.5f * (Tt - st[SCF + i]);
  float psum  = st[SS1 + f] + 0.5f * (Nf - st[SNF + f]);
  float dice  = 1.0f - (2.0f * inter + 1.0f) / (psum + Tt + 1.0f);
  st[SLOSS + i] = bce + dice;
}

// ---------------- greedy matching (Python iterate-while-removing semantics) --------
__global__ void k_match(const float* arenaF, const int* arenaI, float* out) {
  if (threadIdx.x != 0 || blockIdx.x != 0) return;

  const float Nf = (float)NPIX;
  float bce  = arenaF[OFF_BASES + 0] / Nf;
  float dice = 1.0f - (2.0f * arenaF[OFF_BASES + 2] + 1.0f) /
                      (arenaF[OFF_BASES + 1] + arenaF[OFF_BASES + 3] + 1.0f);
  float res = bce + dice;

  int tl[NT]; int tn = 0;
  for (int t = 0; t < NT; ++t)
    if (arenaI[OFF_TCNT + t] > 0) tl[tn++] = t;

  int pfl = arenaI[OFF_PFLAG];
  for (int v = 1; v <= 2; ++v) {
    if (!(pfl & (1 << v))) continue;
    const float* st = arenaF + OFF_V0 + (size_t)(v - 1) * VSTR;
    int NC = arenaI[OFF_NCNT + (v - 1)];
    if (NC > MAXC - 1) NC = MAXC - 1;

    bool alive[MAXC];
    int curCount = 0;
    alive[0] = (st[SNF + 0] > 0.5f);           // background pseudo-component
    if (alive[0]) curCount++;
    for (int f = 1; f <= NC; ++f) { alive[f] = true; curCount++; }

    int i = 0;
    while (i < tn) {
      int t = tl[i];
      if (curCount > 0) {
        float best = 3.4e38f; int mf = -1;
        for (int f = 0; f <= NC; ++f) {
          if (!alive[f]) continue;
          float l = st[SLOSS + f * NT + t];
          if (l < best) { best = l; mf = f; }
        }
        res += best;
        alive[mf] = false; curCount--;
        for (int j = i; j < tn - 1; ++j) tl[j] = tl[j + 1];  // remove current t
        tn--;
      }
      i++;  // Python for-loop advance (skips the element shifted into slot i)
    }
    res += (float)curCount;  // leftover components
  }
  res += (float)tn;          // leftover targets
  out[0] = res;
}

// ---------------- launch ----------------
extern "C" void kernel_launch(void* const* d_in, const int* in_sizes, int n_in,
                              void* d_out, int out_size, void* d_ws, size_t ws_size,
                              hipStream_t stream) {
  const float* pred = (const float*)d_in[0];
  const int*   tgt  = (const int*)d_in[1];

  uint8_t* ws     = (uint8_t*)d_ws;
  uint8_t* pm     = ws + WS_PM;
  int*     labels = (int*)(ws + WS_LABELS);
  int*     cid    = (int*)(ws + WS_CID);
  float*   arenaF = (float*)(ws + WS_ARENA);
  int*     arenaI = (int*)arenaF;

  const int TPB = 256;
  const int PIXBLK = NPIX / TPB;  // 4096

  k_zero<<<(ARENA_DW + TPB - 1) / TPB, TPB, 0, stream>>>(arenaF);
  k_argmax_base<<<512, TPB, 0, stream>>>(pred, tgt, pm, arenaF, arenaI);

  for (int v = 1; v <= 2; ++v) {
    float* st = arenaF + OFF_V0 + (size_t)(v - 1) * VSTR;
    k_cc_init<<<PIXBLK, TPB, 0, stream>>>(pm, labels, v);
    k_cc_merge<<<PIXBLK, TPB, 0, stream>>>(pm, labels, v);
    k_cc_flatten<<<PIXBLK, TPB, 0, stream>>>(labels);
    k_cc_compact<<<PIXBLK, TPB, 0, stream>>>(pm, labels, cid,
                                             arenaI + OFF_NCNT + (v - 1), v);
    k_stats<<<128, TPB, 0, stream>>>(pred, tgt, pm, labels, cid, v, st);
    k_pair<<<(MAXC * NT + TPB - 1) / TPB, TPB, 0, stream>>>(st, arenaI + OFF_TCNT);
  }
  k_match<<<1, 32, 0, stream>>>(arenaF, arenaI, (float*)d_out);
}